// CompleteTableDetectionSystem_88227218195078
// MI455X (gfx1250) — compile-verified
//
#include <hip/hip_runtime.h>
#include <hip/hip_bf16.h>
#include <stdint.h>

#define BATCH 4
#define NBOX  4096
#define NT    1024
#define CHUNK 256
#define CW    8            // 32-bit words per chunk row (256/32)
#define NMS_THR 0.7f

#define AS_GLOBAL __attribute__((address_space(1)))
#define AS_LDS    __attribute__((address_space(3)))

typedef int v4i __attribute__((ext_vector_type(4)));

// ---------- CDNA5 async global->LDS DMA (gfx1250), with asm fallback ----------
__device__ __forceinline__ void async_gather_b128(const float* __restrict__ gsrc,
                                                  void* ldst) {
#if __has_builtin(__builtin_amdgcn_global_load_async_to_lds_b128)
  __builtin_amdgcn_global_load_async_to_lds_b128(
      (AS_GLOBAL v4i*)(float*)gsrc,
      (AS_LDS v4i*)ldst,
      /*imm offset*/0, /*cpol*/0);
#else
  unsigned lds_off = (unsigned)(unsigned long long)(AS_LDS void*)ldst;
  asm volatile("global_load_async_to_lds_b128 %0, %1, off"
               :: "v"(lds_off), "v"(gsrc) : "memory");
#endif
}

__device__ __forceinline__ void wait_async_zero() {
#if __has_builtin(__builtin_amdgcn_s_wait_asynccnt)
  __builtin_amdgcn_s_wait_asynccnt(0);
#else
  asm volatile("s_wait_asynccnt 0" ::: "memory");
#endif
}

__device__ __forceinline__ bool iou_over(float4 a, float fa, float4 b, float fb) {
  float xx1 = fmaxf(a.x, b.x);
  float yy1 = fmaxf(a.y, b.y);
  float xx2 = fminf(a.z, b.z);
  float yy2 = fminf(a.w, b.w);
  float iw  = fmaxf(xx2 - xx1, 0.0f);
  float ih  = fmaxf(yy2 - yy1, 0.0f);
  float inter = iw * ih;
  float iou = inter / (fa + fb - inter);
  return iou > NMS_THR;
}

// ---------- Kernel 1: chained box decode (anchors -> refined -> final) ----------
__global__ void __launch_bounds__(256)
decode_kernel(const float* __restrict__ anchors,   // (N,4)
              const float* __restrict__ bbr,       // (B,N,4)
              const float* __restrict__ pbr,       // (B,N,4)
              float* __restrict__ boxes_ws,        // (B,N,4) scratch
              float* __restrict__ out)             // (B,N,5)
{
  int gid = blockIdx.x * blockDim.x + threadIdx.x;
  if (gid >= BATCH * NBOX) return;
  int n = gid & (NBOX - 1);

  float4 a  = ((const float4*)anchors)[n];
  float4 d1 = ((const float4*)bbr)[gid];
  float4 d2 = ((const float4*)pbr)[gid];

  float w0  = a.z - a.x;
  float h0  = a.w - a.y;
  float cx0 = a.x + 0.5f * w0;
  float cy0 = a.y + 0.5f * h0;
  float cx1 = fmaf(d1.x, w0, cx0);
  float cy1 = fmaf(d1.y, h0, cy0);
  float w1  = __expf(d1.z) * w0;
  float h1  = __expf(d1.w) * h0;

  float cx2 = fmaf(d2.x, w1, cx1);
  float cy2 = fmaf(d2.y, h1, cy1);
  float w2  = __expf(d2.z) * w1;
  float h2  = __expf(d2.w) * h1;

  float4 fb = make_float4(cx2 - 0.5f * w2, cy2 - 0.5f * h2,
                          cx2 + 0.5f * w2, cy2 + 0.5f * h2);
  ((float4*)boxes_ws)[gid] = fb;

  float* o = out + (size_t)gid * 5;
  o[0] = fb.x; o[1] = fb.y; o[2] = fb.z; o[3] = fb.w;
}

// ---------- Kernel 2: per-image sort + chunked greedy NMS, LDS-resident ----------
__global__ void __launch_bounds__(NT)
nms_kernel(const float* __restrict__ boxes_ws,  // (B,N,4)
           const float* __restrict__ scores,    // (B,N)
           float* __restrict__ out)             // (B,N,5)
{
  const int b   = blockIdx.x;
  const int tid = threadIdx.x;

  __shared__ unsigned long long keys[NBOX];     // 32 KB
  __shared__ float4             box[NBOX];      // 64 KB (sorted)
  __shared__ float              area[NBOX];     // 16 KB
  __shared__ unsigned int       supp[NBOX/32];  // 512 B suppression bitmap
  __shared__ unsigned int       ov[CHUNK * CW]; //  8 KB intra-chunk overlap mask

  // 1) keys: descending score, stable ties by smaller original index
  for (int i = tid; i < NBOX; i += NT) {
    unsigned sb = __float_as_uint(scores[(size_t)b * NBOX + i]); // scores >= 0
    keys[i] = ((unsigned long long)sb << 32) | (unsigned)~(unsigned)i;
  }
  __syncthreads();

  // 2) bitonic sort (descending on 64-bit key)
  for (int k = 2; k <= NBOX; k <<= 1) {
    for (int j = k >> 1; j > 0; j >>= 1) {
      for (int t = tid; t < NBOX; t += NT) {
        int ixj = t ^ j;
        if (ixj > t) {
          unsigned long long va = keys[t];
          unsigned long long vb = keys[ixj];
          bool desc = ((t & k) == 0);
          if ((va < vb) == desc) { keys[t] = vb; keys[ixj] = va; }
        }
      }
      __syncthreads();
    }
  }

  // 3) gather sorted boxes global->LDS via CDNA5 async DMA (per-lane addresses)
  const float* base = boxes_ws + (size_t)b * NBOX * 4;
  for (int i = tid; i < NBOX; i += NT) {
    unsigned orig = ~(unsigned)keys[i];
    async_gather_b128(base + (size_t)orig * 4, &box[i]);
  }
  wait_async_zero();
  __syncthreads();

  for (int i = tid; i < NBOX; i += NT) {
    float4 v = box[i];
    area[i] = (v.z - v.x) * (v.w - v.y);
  }
  if (tid < NBOX / 32) supp[tid] = 0u;
  __syncthreads();

  // 4) chunked exact greedy suppression: 3 barriers per 256-box chunk
  for (int c = 0; c < NBOX; c += CHUNK) {
    // --- Phase A: intra-chunk pairwise overlap mask (no atomics) ---
    for (int task = tid; task < CHUNK * CW; task += NT) {
      int il = task >> 3;        // local row  0..255
      int w  = task & (CW - 1);  // word index 0..7
      int gi = c + il;
      float4 bi = box[gi];
      float  ai = area[gi];
      unsigned m = 0;
      int jbase = w << 5;
      #pragma unroll
      for (int t = 0; t < 32; ++t) {
        int jl = jbase + t;
        if (jl > il) {
          int gj = c + jl;
          if (iou_over(bi, ai, box[gj], area[gj])) m |= (1u << t);
        }
      }
      ov[(il << 3) + w] = m;
    }
    __syncthreads();

    // --- Phase B: sequential greedy inside chunk, one wave32, lockstep ---
    if (tid < 32) {
      const int cw0 = c >> 5;
      for (int il = 0; il < CHUNK; ++il) {
        int gi = c + il;
        unsigned cur = 0;
        if (tid == 0) cur = atomicOr(&supp[gi >> 5], 0u);  // ordered DS read
        cur = __shfl(cur, 0, 32);
        if (((cur >> (gi & 31)) & 1u) == 0u) {             // candidate alive
          if (tid < CW) {
            unsigned m = ov[(il << 3) + tid];
            if (m) atomicOr(&supp[cw0 + tid], m);
          }
        }
      }
    }
    __syncthreads();

    // --- Phase C: accepted chunk members suppress all later boxes ---
    unsigned acc[CW];
    #pragma unroll
    for (int w = 0; w < CW; ++w) acc[w] = supp[(c >> 5) + w];

    for (int j = c + CHUNK + tid; j < NBOX; j += NT) {
      if (((supp[j >> 5] >> (j & 31)) & 1u) != 0u) continue;
      float4 bj = box[j];
      float  aj = area[j];
      bool kill = false;
      #pragma unroll
      for (int w = 0; w < CW; ++w) {
        if (kill) break;
        unsigned aw = ~acc[w];                 // set bit == accepted member
        while (aw) {
          int t = __builtin_ctz(aw);
          aw &= aw - 1;
          int gi = c + (w << 5) + t;
          if (iou_over(box[gi], area[gi], bj, aj)) { kill = true; break; }
        }
      }
      if (kill) atomicOr(&supp[j >> 5], 1u << (j & 31));
    }
    __syncthreads();
  }

  // 5) scatter kept scores back to original order (column 4)
  for (int i = tid; i < NBOX; i += NT) {
    unsigned long long k = keys[i];
    unsigned orig = ~(unsigned)k;
    float s = __uint_as_float((unsigned)(k >> 32));
    bool kept = ((supp[i >> 5] >> (i & 31)) & 1u) == 0u;
    out[((size_t)b * NBOX + orig) * 5 + 4] = kept ? s : 0.0f;
  }
}

extern "C" void kernel_launch(void* const* d_in, const int* in_sizes, int n_in,
                              void* d_out, int out_size, void* d_ws, size_t ws_size,
                              hipStream_t stream) {
  const float* anchors = (const float*)d_in[0]; // (N,4)
  const float* bbr     = (const float*)d_in[1]; // (B,N,4)
  const float* pbr     = (const float*)d_in[2]; // (B,N,4)
  const float* scores  = (const float*)d_in[3]; // (B,N)
  float* out      = (float*)d_out;              // (B,N,5)
  float* boxes_ws = (float*)d_ws;               // (B,N,4) = 256 KB scratch

  decode_kernel<<<(BATCH * NBOX) / 256, 256, 0, stream>>>(
      anchors, bbr, pbr, boxes_ws, out);
  nms_kernel<<<BATCH, NT, 0, stream>>>(boxes_ws, scores, out);
}